// LightconvLayer_43808666419738
// MI455X (gfx1250) — compile-verified
//
#include <hip/hip_runtime.h>

// LightConv (depthwise temporal conv, softmax weights) for MI455X / gfx1250.
// Map the 31-tap causal depthwise conv to banded-Toeplitz matmuls on the
// CDNA5 matrix pipe with V_WMMA_F32_16X16X4_F32 (exact fp32):
//   D(16 time x 16 ch) = A(16 x 48 banded softmax weights) * B(48 x 16 input)
// 12 WMMA K=4 chunks per output tile; each wave strip-mines 8 consecutive
// time tiles reusing A (head-invariant) and 8/12 of the B chunks per step.
// Edge strips (causal left pad / right overrun of zero-A columns) use a
// branch-free clamped load path; interior strips use unconditional loads
// from one per-lane base pointer with immediate row offsets.

#define Tn   2048
#define Bn   8
#define Cn   1024
#define Hn   16
#define Kn   31
#define PADL 30

#define NT  8                // time tiles per wave strip
#define WPB 4                // waves per block

typedef __attribute__((ext_vector_type(2))) float v2f;
typedef __attribute__((ext_vector_type(8))) float v8f;

// ---------------------------------------------------------------------------
// Kernel 1: per-head softmax over the 31 kernel taps -> d_ws (16*31 floats).
// ---------------------------------------------------------------------------
__global__ __launch_bounds__(Hn * 32) void lc_softmax(const float* __restrict__ w,
                                                      float* __restrict__ wsm) {
    const int h = threadIdx.x >> 5;   // head = wave id
    const int k = threadIdx.x & 31;   // lane = tap
    float v = (k < Kn) ? w[h * Kn + k] : -__builtin_inff();
    float m = v;
#pragma unroll
    for (int off = 16; off > 0; off >>= 1) m = fmaxf(m, __shfl_xor(m, off, 32));
    float e = (k < Kn) ? __expf(v - m) : 0.0f;
    float s = e;
#pragma unroll
    for (int off = 16; off > 0; off >>= 1) s += __shfl_xor(s, off, 32);
    if (k < Kn) wsm[h * Kn + k] = e / s;
}

// ---------------------------------------------------------------------------
// Strip body. SAFE=true for edge strips (clamped, branch-free loads);
// SAFE=false for interior strips (unconditional loads, immediate offsets).
// ---------------------------------------------------------------------------
template <bool SAFE>
__device__ __forceinline__ void conv_strip(const float* __restrict__ x,
                                           const float* __restrict__ wsm,
                                           float* __restrict__ out,
                                           int h, int t_base, long colBase,
                                           int mlo, int hh) {
    constexpr long rowStride = (long)Bn * Cn;

    // --- A fragments: A[m,s] = wsm[s-m] for 0 <= s-m <= 30, s = 4j + v + 2*hh.
    float a[12][2];
#pragma unroll
    for (int j = 0; j < 12; ++j) {
#pragma unroll
        for (int v = 0; v < 2; ++v) {
            const int s   = 4 * j + v + 2 * hh;
            const int idx = s - mlo;
            a[j][v] = (idx >= 0 && idx < Kn) ? wsm[h * Kn + idx] : 0.0f;
        }
    }

    // Per-lane base: time tt = tBase0 + ofs, ofs a compile-time row offset.
    const int    tBase0 = t_base - PADL + 2 * hh;
    const float* xb     = x + (long)tBase0 * rowStride + colBase;

    auto loadx = [&](int ofs) -> float {
        if (!SAFE) {
            return xb[(long)ofs * rowStride];          // global_load offset:ofs*32768
        } else {
            const int tt  = tBase0 + ofs;
            int       ttc = tt < 0 ? 0 : tt;           // branch-free clamp
            ttc           = ttc > (Tn - 1) ? (Tn - 1) : ttc;
            const float v = x[(long)ttc * rowStride + colBase];
            return ((unsigned)tt < (unsigned)Tn) ? v : 0.0f;  // v_cndmask
        }
    };

    // --- B chunk ring: slot j holds rows s = 4*gci .. 4*gci+3 (gci = 4i + j).
    float bx[12][2];
#pragma unroll
    for (int j = 0; j < 12; ++j) {
#pragma unroll
        for (int v = 0; v < 2; ++v) bx[j][v] = loadx(4 * j + v);
    }

    float* outb = out + (long)(t_base + 8 * hh) * rowStride + colBase;

#pragma unroll
    for (int i = 0; i < NT; ++i) {
        v8f acc = {};
#pragma unroll
        for (int j = 0; j < 12; ++j) {
            v2f av = {a[j][0], a[j][1]};
            v2f bv = {bx[j][0], bx[j][1]};
            // 8 args: (neg_a, A, neg_b, B, c_mod, C, reuse_a, reuse_b)
            acc = __builtin_amdgcn_wmma_f32_16x16x4_f32(
                false, av, false, bv, (short)0, acc, false, false);
        }

        // D: VGPR r, half hh -> time row t_base + 16i + r + 8*hh; streamed (NT).
#pragma unroll
        for (int r = 0; r < 8; ++r)
            __builtin_nontemporal_store(acc[r], &outb[(long)(16 * i + r) * rowStride]);

        // Slide window by one tile = 4 chunks.
        if (i + 1 < NT) {
#pragma unroll
            for (int j = 0; j < 8; ++j) {
                bx[j][0] = bx[j + 4][0];
                bx[j][1] = bx[j + 4][1];
            }
#pragma unroll
            for (int j = 8; j < 12; ++j) {
#pragma unroll
                for (int v = 0; v < 2; ++v)
                    bx[j][v] = loadx(16 * (i + 1) + 4 * j + v);
            }
        }
    }
}

// ---------------------------------------------------------------------------
// Kernel 2: wave task = (batch b, 16-channel tile, strip of NT time tiles).
// ---------------------------------------------------------------------------
__global__ __launch_bounds__(WPB * 32) void lc_conv(const float* __restrict__ x,
                                                    const float* __restrict__ wsm,
                                                    float* __restrict__ out) {
    const int lane = threadIdx.x & 31;
    const int wid  = blockIdx.x * WPB + (threadIdx.x >> 5);

    constexpr int strips = Tn / (16 * NT);   // 16 strips along T
    constexpr int ctiles = Cn / 16;          // 64 channel tiles
    const int strip = wid % strips;
    const int ctile = (wid / strips) % ctiles;
    const int b     = wid / (strips * ctiles);

    const int h      = ctile / ((Cn / Hn) / 16);  // 4 channel tiles per head
    const int c0     = ctile * 16;
    const int t_base = strip * (16 * NT);

    const int  mlo     = lane & 15;   // M for A rows, N for B/D columns
    const int  hh      = lane >> 4;   // lane half
    const long colBase = (long)b * Cn + c0 + mlo;

    // strip is wave-uniform -> uniform branch, no divergence.
    if (strip == 0 || strip == strips - 1)
        conv_strip<true>(x, wsm, out, h, t_base, colBase, mlo, hh);
    else
        conv_strip<false>(x, wsm, out, h, t_base, colBase, mlo, hh);
}

// ---------------------------------------------------------------------------
extern "C" void kernel_launch(void* const* d_in, const int* in_sizes, int n_in,
                              void* d_out, int out_size, void* d_ws, size_t ws_size,
                              hipStream_t stream) {
    const float* x = (const float*)d_in[0];   // (T, B, C) fp32
    const float* w = (const float*)d_in[1];   // (H, K) fp32
    float* out = (float*)d_out;               // (T, B, C) fp32
    float* wsm = (float*)d_ws;                // H*K softmaxed weights

    lc_softmax<<<1, Hn * 32, 0, stream>>>(w, wsm);

    constexpr int total_waves = Bn * (Cn / 16) * (Tn / (16 * NT));  // 8192
    lc_conv<<<total_waves / WPB, WPB * 32, 0, stream>>>(x, wsm, out);
}